// GATEncoder_8899172237586
// MI455X (gfx1250) — compile-verified
//
#include <hip/hip_runtime.h>
#include <math.h>

// ---------------------------------------------------------------------------
// GAT encoder: 3 band GATConvs -> concat -> fused GATConv -> final GATConv,
// each followed by BatchNorm(train stats)+ReLU.  Dense projections use
// CDNA5 V_WMMA_F32_16X16X4_F32 (full fp32 matrix core, exact vs reference).
// Edge softmax / aggregation done with global float atomics (bandwidth bound).
// ---------------------------------------------------------------------------

#define N_NODES 100000
#define N_EDGES 1600000
#define C_IN    128
#define C_HID   128
#define LEAKY_SLOPE 0.2f
#define BN_EPS_ 1e-5f
#define INF_F 3.402823466e38f

typedef __attribute__((ext_vector_type(2))) float v2f;
typedef __attribute__((ext_vector_type(8))) float v8f;

// Only check for the builtin during the actual device compilation pass;
// the host pass of hipcc must not evaluate amdgcn __has_builtin.
#if defined(__HIP_DEVICE_COMPILE__)
#if !__has_builtin(__builtin_amdgcn_wmma_f32_16x16x4_f32)
#error "toolchain lacks __builtin_amdgcn_wmma_f32_16x16x4_f32 (gfx1250)"
#endif
#endif

// ---------------- fp32 WMMA GEMM: H[M,Nc] = X[M,K] * W[K,Nc], row-major ----
// One wave32 computes one 16x16 tile of H.  M,K,Nc all multiples of 16/4 here,
// so every wave that reaches the WMMA has full EXEC (ISA requirement).
__global__ void __launch_bounds__(256)
gemm_wmma_f32(const float* __restrict__ X, const float* __restrict__ W,
              float* __restrict__ H, int M, int K, int Nc) {
  const int lane = threadIdx.x & 31;
  const int wave = threadIdx.x >> 5;
  const int tile = blockIdx.x * (blockDim.x >> 5) + wave;
  const int tilesN = Nc >> 4;
  const int totalTiles = (M >> 4) * tilesN;
  if (tile >= totalTiles) return;           // wave-uniform branch
  const int tm = tile / tilesN;
  const int tn = tile % tilesN;
  const int r  = lane & 15;                 // row m (for A) / col n (for B)
  const int kh = lane >> 4;                 // k-pair select: +0 or +2
  const float* __restrict__ xrow = X + (size_t)(tm * 16 + r) * K;
  const int ncol = tn * 16 + r;
  v8f acc = {};
  for (int k0 = 0; k0 < K; k0 += 4) {
    const int ka = k0 + kh * 2;
    // A (16x4 f32): lane L -> m=L%16, k=(L/16)*2+{0,1}  (contiguous pair)
    v2f a = *(const v2f*)(xrow + ka);
    // B (4x16 f32): lane L -> n=L%16, same k mapping
    v2f b;
    b.x = W[(size_t)(ka    ) * Nc + ncol];
    b.y = W[(size_t)(ka + 1) * Nc + ncol];
    // speculative prefetch of next B k-slice (global_prefetch_b8)
    __builtin_prefetch(W + (size_t)(ka + 4) * Nc + ncol, 0, 3);
    acc = __builtin_amdgcn_wmma_f32_16x16x4_f32(false, a, false, b,
                                                (short)0, acc, false, false);
  }
  // C/D layout: VGPR i, lane L -> m = i + (L/16)*8, n = L%16
  float* hb = H + (size_t)(tm * 16) * Nc + tn * 16;
#pragma unroll
  for (int i = 0; i < 8; ++i)
    hb[(size_t)(i + kh * 8) * Nc + r] = acc[i];
}

// ---------------- helpers ---------------------------------------------------
__global__ void fill_f32(float* __restrict__ p, float v, int n) {
  int i = blockIdx.x * blockDim.x + threadIdx.x;
  if (i < n) p[i] = v;
}

__device__ __forceinline__ void atomicMaxF(float* addr, float val) {
  int* ai = (int*)addr;
  int cur = __float_as_int(*addr);
  while (__int_as_float(cur) < val) {
    int prev = atomicCAS(ai, cur, __float_as_int(val));
    if (prev == cur) break;
    cur = prev;
  }
}

// a = h @ a_src ; b = h @ a_dst   (per-node dot products, C_HID = 128)
__global__ void proj_ab(const float* __restrict__ H, const float* __restrict__ asrc,
                        const float* __restrict__ adst, float* __restrict__ A,
                        float* __restrict__ B, int n) {
  int i = blockIdx.x * blockDim.x + threadIdx.x;
  if (i >= n) return;
  const float* h = H + (size_t)i * C_HID;
  float sa = 0.f, sb = 0.f;
#pragma unroll 4
  for (int c = 0; c < C_HID; ++c) {
    float v = h[c];
    sa = fmaf(v, asrc[c], sa);
    sb = fmaf(v, adst[c], sb);
  }
  A[i] = sa;
  B[i] = sb;
}

// edge table: ei[0..E-1]=src, ei[E..2E-1]=dst; t>=E are implicit self-loops.
__device__ __forceinline__ void fetch_edge(const int* __restrict__ ei, int t,
                                           int& s, int& d) {
  if (t < N_EDGES) { s = ei[t]; d = ei[N_EDGES + t]; }
  else             { s = t - N_EDGES; d = s; }
}

__global__ void edge_score_max(const int* __restrict__ ei, const float* __restrict__ A,
                               const float* __restrict__ B, float* __restrict__ Ebuf,
                               float* __restrict__ Mx) {
  int t = blockIdx.x * blockDim.x + threadIdx.x;
  if (t >= N_EDGES + N_NODES) return;
  int s, d; fetch_edge(ei, t, s, d);
  float e = A[s] + B[d];
  e = (e > 0.f) ? e : LEAKY_SLOPE * e;     // leaky_relu
  Ebuf[t] = e;
  atomicMaxF(&Mx[d], e);                    // segment_max over dst
}

__global__ void edge_exp_sum(const int* __restrict__ ei, const float* __restrict__ Mx,
                             float* __restrict__ Ebuf, float* __restrict__ Z) {
  int t = blockIdx.x * blockDim.x + threadIdx.x;
  if (t >= N_EDGES + N_NODES) return;
  int s, d; fetch_edge(ei, t, s, d);
  float ex = expf(Ebuf[t] - Mx[d]);
  Ebuf[t] = ex;
  atomicAdd(&Z[d], ex);                     // segment_sum over dst
}

// out[dst] += (ex / (z[dst]+1e-16)) * h[src]; one thread = (edge, 4 channels)
__global__ void edge_aggregate(const int* __restrict__ ei, const float* __restrict__ Ebuf,
                               const float* __restrict__ Z, const float* __restrict__ H,
                               float* __restrict__ Out) {
  int g = blockIdx.x * blockDim.x + threadIdx.x;
  const int T = N_EDGES + N_NODES;
  if (g >= T * 32) return;
  int t = g >> 5, lane = g & 31;
  int s, d; fetch_edge(ei, t, s, d);
  float w = Ebuf[t] / (Z[d] + 1e-16f);
  const float4 hv = *(const float4*)(H + (size_t)s * C_HID + (lane << 2));
  float* o = Out + (size_t)d * C_HID + (lane << 2);
  atomicAdd(o + 0, w * hv.x);
  atomicAdd(o + 1, w * hv.y);
  atomicAdd(o + 2, w * hv.z);
  atomicAdd(o + 3, w * hv.w);
}

// one-pass batchnorm stats: per-channel sum & sum-of-squares of (agg + bias)
__global__ void bn_stats(const float* __restrict__ Agg, const float* __restrict__ bias,
                         float* __restrict__ Ssum, float* __restrict__ Ssq) {
  const int c = threadIdx.x;                // blockDim.x == 128
  const int r0 = blockIdx.x * 256;
  float s = 0.f, q = 0.f;
  for (int r = 0; r < 256; ++r) {
    int row = r0 + r;
    if (row >= N_NODES) break;
    float v = Agg[(size_t)row * C_HID + c] + bias[c];
    s += v;
    q = fmaf(v, v, q);
  }
  atomicAdd(&Ssum[c], s);
  atomicAdd(&Ssq[c], q);
}

__global__ void bn_apply_relu(const float* __restrict__ Agg, const float* __restrict__ bias,
                              const float* __restrict__ Ssum, const float* __restrict__ Ssq,
                              const float* __restrict__ g, const float* __restrict__ b,
                              float* __restrict__ Out, int ldOut) {
  int i = blockIdx.x * blockDim.x + threadIdx.x;
  if (i >= N_NODES * C_HID) return;
  int row = i >> 7, c = i & 127;
  const float inv_n = 1.0f / (float)N_NODES;
  float mu  = Ssum[c] * inv_n;
  float var = fmaf(Ssq[c], inv_n, -mu * mu);  // biased batch variance
  float v = Agg[i] + bias[c];
  float y = fmaf(g[c] * (v - mu), rsqrtf(var + BN_EPS_), b[c]);
  Out[(size_t)row * ldOut + c] = fmaxf(y, 0.f);
}

// ---------------- one full GATConv + BN + ReLU layer ------------------------
static void run_gat_layer(const float* X, int K, const float* W, const float* asrc,
                          const float* adst, const float* bias, const float* bng,
                          const float* bnb, const int* ei,
                          float* hbuf, float* agg, float* Ab, float* Bb, float* Mx,
                          float* Z, float* Ebuf, float* Ssum, float* Ssq,
                          float* outp, int ldOut, hipStream_t s) {
  const int tiles = (N_NODES / 16) * (C_HID / 16);
  gemm_wmma_f32<<<(tiles + 7) / 8, 256, 0, s>>>(X, W, hbuf, N_NODES, K, C_HID);
  proj_ab<<<(N_NODES + 255) / 256, 256, 0, s>>>(hbuf, asrc, adst, Ab, Bb, N_NODES);
  fill_f32<<<(N_NODES + 255) / 256, 256, 0, s>>>(Mx, -INF_F, N_NODES);
  fill_f32<<<(N_NODES + 255) / 256, 256, 0, s>>>(Z, 0.f, N_NODES);
  fill_f32<<<(N_NODES * C_HID + 255) / 256, 256, 0, s>>>(agg, 0.f, N_NODES * C_HID);
  fill_f32<<<1, 256, 0, s>>>(Ssum, 0.f, C_HID);
  fill_f32<<<1, 256, 0, s>>>(Ssq, 0.f, C_HID);
  const int T = N_EDGES + N_NODES;
  edge_score_max<<<(T + 255) / 256, 256, 0, s>>>(ei, Ab, Bb, Ebuf, Mx);
  edge_exp_sum<<<(T + 255) / 256, 256, 0, s>>>(ei, Mx, Ebuf, Z);
  edge_aggregate<<<(T * 32 + 255) / 256, 256, 0, s>>>(ei, Ebuf, Z, hbuf, agg);
  bn_stats<<<(N_NODES + 255) / 256, 128, 0, s>>>(agg, bias, Ssum, Ssq);
  bn_apply_relu<<<(N_NODES * C_HID + 255) / 256, 256, 0, s>>>(agg, bias, Ssum, Ssq,
                                                              bng, bnb, outp, ldOut);
}

// ---------------- entry -----------------------------------------------------
extern "C" void kernel_launch(void* const* d_in, const int* in_sizes, int n_in,
                              void* d_out, int out_size, void* d_ws, size_t ws_size,
                              hipStream_t stream) {
  (void)in_sizes; (void)n_in; (void)out_size; (void)ws_size;
#define DF(i) ((const float*)d_in[(i)])
  const float* x_alpha = DF(0);
  const float* x_beta  = DF(1);
  const float* x_theta = DF(2);
  const int* ei_alpha = (const int*)d_in[3];
  const int* ei_beta  = (const int*)d_in[4];
  const int* ei_theta = (const int*)d_in[5];
  // params in setup_inputs() insertion order:
  // alpha{W,a_src,a_dst,b}=6..9  beta=10..13  theta=14..17
  // fused=18..21  final=22..25
  // bn_alpha{g,b}=26,27  bn_beta=28,29  bn_theta=30,31  bn_fused=32,33  bn_final=34,35

  float* ws = (float*)d_ws;
  size_t off = 0;
  float* concat = ws + off; off += (size_t)N_NODES * 3 * C_HID; // [N,384]
  float* hbuf   = ws + off; off += (size_t)N_NODES * C_HID;     // h = x@W
  float* agg    = ws + off; off += (size_t)N_NODES * C_HID;     // GAT accum
  float* ofb    = ws + off; off += (size_t)N_NODES * C_HID;     // fused output
  float* Ab   = ws + off; off += N_NODES;
  float* Bb   = ws + off; off += N_NODES;
  float* Mx   = ws + off; off += N_NODES;
  float* Zb   = ws + off; off += N_NODES;
  float* Eb   = ws + off; off += (size_t)(N_EDGES + N_NODES);
  float* Ssum = ws + off; off += C_HID;
  float* Ssq  = ws + off; off += C_HID;

  // three band layers write into column slices of the concat buffer
  run_gat_layer(x_alpha, C_IN, DF(6),  DF(7),  DF(8),  DF(9),  DF(26), DF(27),
                ei_alpha, hbuf, agg, Ab, Bb, Mx, Zb, Eb, Ssum, Ssq,
                concat + 0 * C_HID, 3 * C_HID, stream);
  run_gat_layer(x_beta, C_IN, DF(10), DF(11), DF(12), DF(13), DF(28), DF(29),
                ei_beta, hbuf, agg, Ab, Bb, Mx, Zb, Eb, Ssum, Ssq,
                concat + 1 * C_HID, 3 * C_HID, stream);
  run_gat_layer(x_theta, C_IN, DF(14), DF(15), DF(16), DF(17), DF(30), DF(31),
                ei_theta, hbuf, agg, Ab, Bb, Mx, Zb, Eb, Ssum, Ssq,
                concat + 2 * C_HID, 3 * C_HID, stream);
  // fused layer: [N,384] @ [384,128], edges = alpha band
  run_gat_layer(concat, 3 * C_HID, DF(18), DF(19), DF(20), DF(21), DF(32), DF(33),
                ei_alpha, hbuf, agg, Ab, Bb, Mx, Zb, Eb, Ssum, Ssq,
                ofb, C_HID, stream);
  // final layer -> d_out [N,128]
  run_gat_layer(ofb, C_HID, DF(22), DF(23), DF(24), DF(25), DF(34), DF(35),
                ei_alpha, hbuf, agg, Ab, Bb, Mx, Zb, Eb, Ssum, Ssq,
                (float*)d_out, C_HID, stream);
#undef DF
}